// GAT_Solution_6511170421157
// MI455X (gfx1250) — compile-verified
//
#include <hip/hip_runtime.h>
#include <math.h>

// ---------------- problem constants ----------------
constexpr int NS   = 10;     // num solutions
constexpr int NB   = 32;     // batch
constexpr int NG   = 1000;   // graph size (tour length)
constexpr int NE   = 128;    // embed dim
constexpr int NHD  = 8;      // heads
constexpr int NKD  = 16;     // head dim
constexpr int NMSH = 16;     // mix hidden
constexpr int NTILE = (NG + 15) / 16;  // 63 row-tiles per (s,b)
constexpr int LDSW  = 132;             // padded LDS row stride (floats)
constexpr int NR    = NB * NG;         // 32000 rows

typedef __attribute__((ext_vector_type(2))) float v2f;
typedef __attribute__((ext_vector_type(8))) float v8f;

#define WMMA_F32(a, b, c) \
  __builtin_amdgcn_wmma_f32_16x16x4_f32(false, (a), false, (b), (short)0, (c), false, false)

__device__ inline float sigmoidf_(float x) { return 1.0f / (1.0f + __expf(-x)); }

// ============ Kernel A: per-(s,b,t) edge score via WMMA q/k + mix MLP ============
__global__ __launch_bounds__(32) void edge_cost_kernel(
    const float* __restrict__ node_embed, const int* __restrict__ solutions,
    const float* __restrict__ costs, const float* __restrict__ dist,
    const float* __restrict__ Wq, const float* __restrict__ Wk,
    const float* __restrict__ m1w, const float* __restrict__ m1b,
    const float* __restrict__ m2w, const float* __restrict__ m2b,
    const float* __restrict__ nhw,
    float* __restrict__ edge_w, int* __restrict__ edge_dst) {
  __shared__ float Asolu[16 * LDSW];
  __shared__ float Anext[16 * LDSW];
  __shared__ float dotl[16 * NHD];
  __shared__ float ecl[16];
  __shared__ int srcl[16], dstl[16];

  const int blk  = blockIdx.x;
  const int tile = blk % NTILE;
  const int sb   = blk / NTILE;
  const int b    = sb % NB;
  const int s    = sb / NB;
  const int t0   = tile * 16;
  const int lane = threadIdx.x;

  const int* __restrict__ sol = solutions + (size_t)(s * NB + b) * NG;

  if (lane < 16) {
    int t  = t0 + lane;
    int tv = (t < NG) ? t : 0;
    srcl[lane] = sol[tv];
    dstl[lane] = sol[(tv + 1) % NG];                       // nxt = roll(sol,-1)
  }
  __syncthreads();

  // gather 16 src rows + 16 next rows into LDS, one float4 per lane per row
#pragma unroll 4
  for (int r = 0; r < 16; ++r) {
    const float4* ps = (const float4*)(node_embed + ((size_t)b * NG + srcl[r]) * NE);
    const float4* pn = (const float4*)(node_embed + ((size_t)b * NG + dstl[r]) * NE);
    ((float4*)(Asolu + r * LDSW))[lane] = ps[lane];
    ((float4*)(Anext + r * LDSW))[lane] = pn[lane];
  }
  if (lane < 16)
    ecl[lane] = dist[((size_t)b * NG + srcl[lane]) * NG + dstl[lane]];
  __syncthreads();

  const int m    = lane & 15;
  const int hi   = lane >> 4;
  const int ktop = hi * 2;

  const float* __restrict__ aq_r = Asolu + m * LDSW + ktop;
  const float* __restrict__ ak_r = Anext + m * LDSW + ktop;

  // per head: Qh(16x16) and Kh(16x16) in one fused K-loop with two
  // independent WMMA accumulator chains, then row-dot by shuffle reduce.
  for (int h = 0; h < NHD; ++h) {
    const int n0 = h * NKD;
    const float* __restrict__ wq_r = Wq + (size_t)(n0 + m) * NE + ktop;
    const float* __restrict__ wk_r = Wk + (size_t)(n0 + m) * NE + ktop;
    v8f cq = {0.f, 0.f, 0.f, 0.f, 0.f, 0.f, 0.f, 0.f};
    v8f ck = {0.f, 0.f, 0.f, 0.f, 0.f, 0.f, 0.f, 0.f};
#pragma unroll 8
    for (int k = 0; k < NE; k += 4) {
      v2f aq = *(const v2f*)(aq_r + k);     // A-frag (LDS), even col -> 8B aligned
      v2f ak = *(const v2f*)(ak_r + k);
      v2f bq = *(const v2f*)(wq_r + k);     // B-frag (L2-hot weights)
      v2f bk = *(const v2f*)(wk_r + k);
      cq = WMMA_F32(aq, bq, cq);
      ck = WMMA_F32(ak, bk, ck);
    }
    float pr[8];
#pragma unroll
    for (int v = 0; v < 8; ++v) pr[v] = cq[v] * ck[v];
#pragma unroll
    for (int off = 1; off < 16; off <<= 1) {
#pragma unroll
      for (int v = 0; v < 8; ++v) pr[v] += __shfl_xor(pr[v], off, 32);
    }
    if (m == 0) {
#pragma unroll
      for (int v = 0; v < 8; ++v)
        dotl[(v + 8 * hi) * NHD + h] = pr[v] * 0.0625f;   // (1/sqrt(KD))/sqrt(KD)
    }
  }
  __syncthreads();

  // mix MLP: each lane handles 4 (t,h) pairs; reduce heads via xor-16
  float partial = 0.f;
  const int t = lane & 15;
#pragma unroll
  for (int j = 0; j < 4; ++j) {
    const int h = (lane >> 4) + 2 * j;
    const float d  = dotl[t * NHD + h];
    const float ec = ecl[t];
    float acc2 = m2b[h];
#pragma unroll
    for (int mm = 0; mm < NMSH; ++mm) {
      float v = fmaf(d, m1w[(h * 2 + 0) * NMSH + mm],
                fmaf(ec, m1w[(h * 2 + 1) * NMSH + mm], m1b[h * NMSH + mm]));
      v = fmaxf(v, 0.f);
      acc2 = fmaf(v, m2w[h * NMSH + mm], acc2);
    }
    partial = fmaf(acc2, nhw[h], partial);
  }
  partial += __shfl_xor(partial, 16, 32);

  if (lane < 16) {
    const int tg = t0 + lane;
    if (tg < NG) {
      const float cval = partial / costs[s * NB + b];
      const size_t eidx = ((size_t)b * NG + srcl[lane]) * NS + s;  // write-once: permutation
      edge_w[eidx]   = cval;
      edge_dst[eidx] = dstl[lane];
    }
  }
}

// ============ Kernel B: 10-entry merge + softmax + sparse SpMM ============
__global__ __launch_bounds__(256) void attn_spmm_kernel(
    const float* __restrict__ node_embed, const float* __restrict__ edge_w,
    const int* __restrict__ edge_dst, float* __restrict__ solu_embed) {
  const int wave = threadIdx.x >> 5;
  const int lane = threadIdx.x & 31;
  const int row  = blockIdx.x * 8 + wave;     // (b,src) row
  if (row >= NR) return;
  const int b = row / NG;

  const float* __restrict__ wp = edge_w + (size_t)row * NS;
  const int*   __restrict__ dp = edge_dst + (size_t)row * NS;
  int   d[NS];
  float w[NS];
  bool  dead[NS];
#pragma unroll
  for (int i = 0; i < NS; ++i) { d[i] = dp[i]; w[i] = wp[i]; dead[i] = false; }
  // merge duplicate destination columns (scatter-ADD semantics of reference)
#pragma unroll
  for (int i = 0; i < NS; ++i) {
    if (dead[i]) continue;
#pragma unroll
    for (int j = i + 1; j < NS; ++j)
      if (!dead[j] && d[j] == d[i]) { w[i] += w[j]; dead[j] = true; }
  }
  // softmax over surviving finite entries (exact-zero -> NEG in reference -> prob 0)
  float mx = -INFINITY;
#pragma unroll
  for (int i = 0; i < NS; ++i)
    if (!dead[i] && w[i] != 0.f) mx = fmaxf(mx, w[i]);
  float p[NS], den = 0.f;
#pragma unroll
  for (int i = 0; i < NS; ++i) {
    p[i] = (!dead[i] && w[i] != 0.f) ? __expf(w[i] - mx) : 0.f;
    den += p[i];
  }
  const float rinv = (den > 0.f) ? 1.f / den : 0.f;

  float4 acc = make_float4(0.f, 0.f, 0.f, 0.f);
#pragma unroll
  for (int i = 0; i < NS; ++i) {
    if (p[i] != 0.f) {
      const float pi = p[i] * rinv;
      float4 ne = ((const float4*)(node_embed + ((size_t)b * NG + d[i]) * NE))[lane];
      acc.x = fmaf(pi, ne.x, acc.x);
      acc.y = fmaf(pi, ne.y, acc.y);
      acc.z = fmaf(pi, ne.z, acc.z);
      acc.w = fmaf(pi, ne.w, acc.w);
    }
  }
  ((float4*)(solu_embed + (size_t)row * NE))[lane] = acc;
}

// ============ Kernel C: GRU cell via fused 6-chain WMMA + ELU epilogue ============
__global__ __launch_bounds__(32) void gru_kernel(
    const float* __restrict__ solu_embed, const float* __restrict__ hold,
    const float* __restrict__ wih, const float* __restrict__ whh,
    const float* __restrict__ bih, const float* __restrict__ bhh,
    float* __restrict__ out) {
  __shared__ float X[16 * LDSW];
  __shared__ float Hh[16 * LDSW];
  const int lane = threadIdx.x;
  const size_t r0 = (size_t)blockIdx.x * 16;

#pragma unroll 4
  for (int r = 0; r < 16; ++r) {
    ((float4*)(X  + r * LDSW))[lane] = ((const float4*)(solu_embed + (r0 + r) * NE))[lane];
    ((float4*)(Hh + r * LDSW))[lane] = ((const float4*)(hold       + (r0 + r) * NE))[lane];
  }
  __syncthreads();

  const int m    = lane & 15;
  const int hi   = lane >> 4;
  const int ktop = hi * 2;

  const float* __restrict__ ax_r = X  + m * LDSW + ktop;
  const float* __restrict__ ah_r = Hh + m * LDSW + ktop;

  for (int c0 = 0; c0 < NE; c0 += 16) {
    // six gate GEMM tiles in one fused K-loop: A-frags shared across 3 gates,
    // six independent WMMA accumulator chains for maximum pipeline overlap.
    const float* __restrict__ wir = wih + (size_t)(c0 + m) * NE + ktop;
    const float* __restrict__ wiz = wih + (size_t)(NE + c0 + m) * NE + ktop;
    const float* __restrict__ win = wih + (size_t)(2 * NE + c0 + m) * NE + ktop;
    const float* __restrict__ whr = whh + (size_t)(c0 + m) * NE + ktop;
    const float* __restrict__ whz = whh + (size_t)(NE + c0 + m) * NE + ktop;
    const float* __restrict__ whn = whh + (size_t)(2 * NE + c0 + m) * NE + ktop;

    v8f ir  = {0.f, 0.f, 0.f, 0.f, 0.f, 0.f, 0.f, 0.f};
    v8f iz  = ir, in_ = ir, hr = ir, hz = ir, hn = ir;
#pragma unroll 4
    for (int k = 0; k < NE; k += 4) {
      v2f ax = *(const v2f*)(ax_r + k);
      v2f ah = *(const v2f*)(ah_r + k);
      ir  = WMMA_F32(ax, *(const v2f*)(wir + k), ir);
      iz  = WMMA_F32(ax, *(const v2f*)(wiz + k), iz);
      in_ = WMMA_F32(ax, *(const v2f*)(win + k), in_);
      hr  = WMMA_F32(ah, *(const v2f*)(whr + k), hr);
      hz  = WMMA_F32(ah, *(const v2f*)(whz + k), hz);
      hn  = WMMA_F32(ah, *(const v2f*)(whn + k), hn);
    }

    const int n = c0 + m;                    // output column
    const float bir = bih[n], biz = bih[NE + n], bin = bih[2 * NE + n];
    const float bhr = bhh[n], bhz = bhh[NE + n], bhn = bhh[2 * NE + n];
#pragma unroll
    for (int v = 0; v < 8; ++v) {
      const int M = v + 8 * hi;
      const float hval = Hh[M * LDSW + n];
      const float rg = sigmoidf_((ir[v] + bir) + (hr[v] + bhr));
      const float zg = sigmoidf_((iz[v] + biz) + (hz[v] + bhz));
      const float ng = tanhf((in_[v] + bin) + rg * (hn[v] + bhn));
      const float nw = (1.f - zg) * ng + zg * hval;
      const size_t o = (r0 + M) * NE + n;
      out[(size_t)NR * NE + o] = nw;                              // output[1] = new
      out[o] = (nw > 0.f) ? nw : (__expf(nw) - 1.f);              // output[0] = elu(new)
    }
  }
}

// ============ host launch ============
extern "C" void kernel_launch(void* const* d_in, const int* in_sizes, int n_in,
                              void* d_out, int out_size, void* d_ws, size_t ws_size,
                              hipStream_t stream) {
  (void)in_sizes; (void)n_in; (void)out_size; (void)ws_size;
  const float* node_embed = (const float*)d_in[0];
  const int*   solutions  = (const int*)d_in[1];
  const float* costs      = (const float*)d_in[2];
  const float* dist       = (const float*)d_in[3];
  const float* hold       = (const float*)d_in[4];
  const float* Wq         = (const float*)d_in[5];
  const float* Wk         = (const float*)d_in[6];
  const float* m1w        = (const float*)d_in[7];
  const float* m1b        = (const float*)d_in[8];
  const float* m2w        = (const float*)d_in[9];
  const float* m2b        = (const float*)d_in[10];
  const float* nhw        = (const float*)d_in[11];
  const float* wih        = (const float*)d_in[12];
  const float* whh        = (const float*)d_in[13];
  const float* bih        = (const float*)d_in[14];
  const float* bhh        = (const float*)d_in[15];
  float* out = (float*)d_out;

  char*  ws         = (char*)d_ws;
  float* edge_w     = (float*)ws;                                    // NB*NG*NS f32
  int*   edge_dst   = (int*)(ws + sizeof(float) * NB * NG * NS);     // NB*NG*NS i32
  float* solu_embed = (float*)(ws + 2 * sizeof(float) * NB * NG * NS); // NB*NG*NE f32

  edge_cost_kernel<<<dim3(NS * NB * NTILE), dim3(32), 0, stream>>>(
      node_embed, solutions, costs, dist, Wq, Wk, m1w, m1b, m2w, m2b, nhw,
      edge_w, edge_dst);
  attn_spmm_kernel<<<dim3(NR / 8), dim3(256), 0, stream>>>(
      node_embed, edge_w, edge_dst, solu_embed);
  gru_kernel<<<dim3(NR / 16), dim3(32), 0, stream>>>(
      solu_embed, hold, wih, whh, bih, bhh, out);
}